// BPNN_24129126269307
// MI455X (gfx1250) — compile-verified
//
#include <hip/hip_runtime.h>
#include <hip/hip_bf16.h>

typedef float v2f __attribute__((ext_vector_type(2)));
typedef float v8f __attribute__((ext_vector_type(8)));

#define NATOM   60000
#define FPD     120
#define HID     512
#define NELEM   3
#define NIMG    600
#define NNZV    5000000
#define NFORCEV 180000
#define TILE    32                 // atoms per workgroup
#define NTILES  (NATOM / TILE)     // 1875 (exact)
#define LDFP    122                // padded LDS stride (even -> 8B aligned, conflict-free)
#define LDH     516                // padded LDS stride for 512-wide activations

// ---------------------------------------------------------------------------
// CDNA5 fp32 WMMA: D(16x16) = A(16x4) * B(4x16) + C
// ---------------------------------------------------------------------------
__device__ __forceinline__ v8f wmma_f32(v2f a, v2f b, v8f c) {
  return __builtin_amdgcn_wmma_f32_16x16x4_f32(false, a, false, b, (short)0, c,
                                               false, false);
}

// Native V_TANH_F32 (gfx1250 TRANS op) if the toolchain exposes it; the TRANS
// pipe co-executes with WMMA, and this avoids the exp-based branchy expansion.
#if defined(__has_builtin)
#if __has_builtin(__builtin_amdgcn_tanhf)
#define FAST_TANH(x) __builtin_amdgcn_tanhf(x)
#endif
#endif
#ifndef FAST_TANH
#define FAST_TANH(x) tanhf(x)
#endif

__device__ __forceinline__ int classify(int an) {
  // unique_atoms = [1, 8, 29]
  return (an == 1) ? 0 : ((an == 8) ? 1 : 2);
}

// ---------------------------------------------------------------------------
// small helper kernels
// ---------------------------------------------------------------------------
__global__ void k_zero_f32(float* p, int n) {
  int i = blockIdx.x * 256 + threadIdx.x;
  if (i < n) p[i] = 0.0f;
}
__global__ void k_zero_i32(int* p, int n) {
  int i = threadIdx.x;
  if (i < n) p[i] = 0;
}
__global__ void k_hist(const int* __restrict__ an, int* cnt) {
  int i = blockIdx.x * 256 + threadIdx.x;
  if (i < NATOM) atomicAdd(&cnt[classify(an[i])], 1);
}
__global__ void k_scan(const int* cnt, int* cur) {
  if (threadIdx.x == 0) {
    cur[0] = 0;
    cur[1] = cnt[0];
    cur[2] = cnt[0] + cnt[1];
  }
}
__global__ void k_scatter(const int* __restrict__ an, int* cur,
                          int* __restrict__ perm) {
  int i = blockIdx.x * 256 + threadIdx.x;
  if (i < NATOM) {
    int pos = atomicAdd(&cur[classify(an[i])], 1);
    perm[pos] = i;
  }
}
// forces = -fprimes^T @ g  (one atomic per nnz; 720KB target stays in L2)
__global__ void k_force(const int* __restrict__ rows, const int* __restrict__ cols,
                        const float* __restrict__ vals, const float* __restrict__ g,
                        float* __restrict__ forces) {
  int t = blockIdx.x * 256 + threadIdx.x;
  if (t < NNZV) atomicAdd(&forces[cols[t]], -vals[t] * g[rows[t]]);
}

// ---------------------------------------------------------------------------
// forward layer: out = tanh(A @ W + b), A is [32 x K] in LDS, W is [K x 512]
// wave w owns N-tiles 4w..4w+3 for both 16-row M subtiles (B reused across M)
// ---------------------------------------------------------------------------
__device__ __forceinline__ void fwd_layer(const float* __restrict__ aS, int lda,
                                          const float* __restrict__ W,
                                          const float* __restrict__ biasS,
                                          float* __restrict__ outS, int K,
                                          int wave, int lane) {
  const int ncl   = lane & 15;
  const int akoff = (lane >> 4) << 1;   // hi half-wave holds K+2,K+3
  v8f acc[2][4];
#pragma unroll
  for (int nti = 0; nti < 4; ++nti) {
    float bv = biasS[(wave * 4 + nti) * 16 + ncl];
    v8f c;
#pragma unroll
    for (int i = 0; i < 8; ++i) c[i] = bv;
    acc[0][nti] = c;
    acc[1][nti] = c;
  }
  for (int k0 = 0; k0 < K; k0 += 4) {
    const int kr = k0 + akoff;
    v2f a0 = *(const v2f*)(aS + ncl * lda + kr);
    v2f a1 = *(const v2f*)(aS + (ncl + 16) * lda + kr);
#pragma unroll
    for (int nti = 0; nti < 4; ++nti) {
      int col = (wave * 4 + nti) * 16 + ncl;
      v2f b;
      b.x = W[kr * HID + col];
      b.y = W[(kr + 1) * HID + col];
      acc[0][nti] = wmma_f32(a0, b, acc[0][nti]);
      acc[1][nti] = wmma_f32(a1, b, acc[1][nti]);
    }
  }
  const int rb = (lane >> 4) << 3;      // C layout: hi half-wave = rows+8
#pragma unroll
  for (int m = 0; m < 2; ++m)
#pragma unroll
    for (int nti = 0; nti < 4; ++nti) {
      int nb = (wave * 4 + nti) * 16;
      v8f c = acc[m][nti];
#pragma unroll
      for (int i = 0; i < 8; ++i)
        outS[(m * 16 + rb + i) * LDH + nb + ncl] = FAST_TANH(c[i]);
    }
}

// ---------------------------------------------------------------------------
// main fused fwd+bwd MLP kernel: one 32-atom (element-sorted) tile per block
// ---------------------------------------------------------------------------
__global__ __launch_bounds__(256) void bpnn_mlp(
    const float* __restrict__ FPg, const int* __restrict__ AN,
    const int* __restrict__ IMG, const float* __restrict__ W1,
    const float* __restrict__ B1, const float* __restrict__ W2,
    const float* __restrict__ B2, const float* __restrict__ W3,
    const float* __restrict__ B3, const int* __restrict__ perm,
    float* __restrict__ g, float* __restrict__ energy) {
  extern __shared__ float smem[];
  float* fpS = smem;                     // 32 x 122
  float* h1S = fpS + TILE * LDFP;        // 32 x 516
  float* h2S = h1S + TILE * LDH;         // 32 x 516
  float* dhS = h2S + TILE * LDH;         // 32 x 516
  float* b1S = dhS + TILE * LDH;         // 512
  float* b2S = b1S + HID;                // 512
  float* w3S = b2S + HID;                // 512
  float* oS  = w3S + HID;                // 32
  int* permS = (int*)(oS + TILE);        // 32
  int* elemS = permS + TILE;             // 32
  int* imgS  = elemS + TILE;             // 32

  const int tid  = threadIdx.x;
  const int lane = tid & 31;
  const int wave = tid >> 5;
  const int base = blockIdx.x * TILE;

  if (tid < TILE) {
    int p = perm[base + tid];
    permS[tid] = p;
    elemS[tid] = classify(AN[p]);
    imgS[tid]  = IMG[p];
  }
  __syncthreads();
  // gather permuted fingerprint rows into LDS
  for (int idx = tid; idx < TILE * FPD; idx += 256) {
    int r = idx / FPD, c = idx % FPD;
    fpS[r * LDFP + c] = FPg[(size_t)permS[r] * FPD + c];
  }
  __syncthreads();

  const int e0 = elemS[0], e1 = elemS[TILE - 1];  // sorted: usually e0==e1

  for (int e = e0; e <= e1; ++e) {
    const float* W1e = W1 + (size_t)e * FPD * HID;
    const float* W2e = W2 + (size_t)e * HID * HID;
    for (int i = tid; i < HID; i += 256) {
      b1S[i] = B1[e * HID + i];
      b2S[i] = B2[e * HID + i];
      w3S[i] = W3[e * HID + i];
    }
    __syncthreads();

    // ----- layer 1: h1 = tanh(fp @ W1 + b1) -----
    fwd_layer(fpS, LDFP, W1e, b1S, h1S, FPD, wave, lane);
    __syncthreads();
    // ----- layer 2: h2 = tanh(h1 @ W2 + b2) -----
    fwd_layer(h1S, LDH, W2e, b2S, h2S, HID, wave, lane);
    __syncthreads();

    // ----- layer 3 + per-image energy: o = mask*(h2.W3 + b3) -----
    if (tid < TILE) oS[tid] = 0.0f;
    __syncthreads();
    {
      int a = tid >> 3, seg = tid & 7;
      float s = 0.0f;
      for (int k = seg * 64; k < seg * 64 + 64; ++k)
        s += h2S[a * LDH + k] * w3S[k];
      atomicAdd(&oS[a], s);
    }
    __syncthreads();
    if (tid < TILE && elemS[tid] == e)
      atomicAdd(&energy[imgS[tid]], oS[tid] + B3[e]);
    __syncthreads();

    // ----- bwd2: dh1 = (mask*W3*(1-h2^2)) @ W2^T -----
    {
      const int ncl   = lane & 15;
      const int akoff = (lane >> 4) << 1;
      const float m0  = (elemS[ncl] == e) ? 1.0f : 0.0f;
      const float m1  = (elemS[ncl + 16] == e) ? 1.0f : 0.0f;
      v8f acc[2][4];
      v8f z;
#pragma unroll
      for (int i = 0; i < 8; ++i) z[i] = 0.0f;
#pragma unroll
      for (int nti = 0; nti < 4; ++nti) { acc[0][nti] = z; acc[1][nti] = z; }
      for (int k0 = 0; k0 < HID; k0 += 4) {
        const int kr = k0 + akoff;
        v2f w3v = *(const v2f*)(w3S + kr);
        v2f t0  = *(const v2f*)(h2S + ncl * LDH + kr);
        v2f t1  = *(const v2f*)(h2S + (ncl + 16) * LDH + kr);
        v2f a0, a1;
        a0.x = m0 * w3v.x * (1.0f - t0.x * t0.x);
        a0.y = m0 * w3v.y * (1.0f - t0.y * t0.y);
        a1.x = m1 * w3v.x * (1.0f - t1.x * t1.x);
        a1.y = m1 * w3v.y * (1.0f - t1.y * t1.y);
#pragma unroll
        for (int nti = 0; nti < 4; ++nti) {
          int col = (wave * 4 + nti) * 16 + ncl;
          v2f b = *(const v2f*)(W2e + (size_t)col * HID + kr);  // W2^T: contiguous
          acc[0][nti] = wmma_f32(a0, b, acc[0][nti]);
          acc[1][nti] = wmma_f32(a1, b, acc[1][nti]);
        }
      }
      const int rb = (lane >> 4) << 3;
#pragma unroll
      for (int m = 0; m < 2; ++m)
#pragma unroll
        for (int nti = 0; nti < 4; ++nti) {
          int nb = (wave * 4 + nti) * 16;
          v8f c = acc[m][nti];
#pragma unroll
          for (int i = 0; i < 8; ++i)
            dhS[(m * 16 + rb + i) * LDH + nb + ncl] = c[i];
        }
    }
    __syncthreads();

    // ----- bwd1: g = (dh1*(1-h1^2)) @ W1^T ; wave w owns cols 16w..16w+15 -----
    {
      const int ncl   = lane & 15;
      const int akoff = (lane >> 4) << 1;
      const int col   = wave * 16 + ncl;
      const float cok = (col < FPD) ? 1.0f : 0.0f;
      const int colc  = (col < FPD) ? col : 0;
      v8f acc[2];
#pragma unroll
      for (int i = 0; i < 8; ++i) { acc[0][i] = 0.0f; acc[1][i] = 0.0f; }
      for (int k0 = 0; k0 < HID; k0 += 4) {
        const int kr = k0 + akoff;
        v2f d0 = *(const v2f*)(dhS + ncl * LDH + kr);
        v2f d1 = *(const v2f*)(dhS + (ncl + 16) * LDH + kr);
        v2f t0 = *(const v2f*)(h1S + ncl * LDH + kr);
        v2f t1 = *(const v2f*)(h1S + (ncl + 16) * LDH + kr);
        v2f a0, a1;
        a0.x = d0.x * (1.0f - t0.x * t0.x);
        a0.y = d0.y * (1.0f - t0.y * t0.y);
        a1.x = d1.x * (1.0f - t1.x * t1.x);
        a1.y = d1.y * (1.0f - t1.y * t1.y);
        v2f b = *(const v2f*)(W1e + (size_t)colc * HID + kr);  // W1^T: contiguous
        b.x *= cok;
        b.y *= cok;
        acc[0] = wmma_f32(a0, b, acc[0]);
        acc[1] = wmma_f32(a1, b, acc[1]);
      }
      const int rb = (lane >> 4) << 3;
      if (col < FPD) {
#pragma unroll
        for (int m = 0; m < 2; ++m) {
          v8f c = acc[m];
#pragma unroll
          for (int i = 0; i < 8; ++i) {
            int lr = m * 16 + rb + i;
            if (elemS[lr] == e)  // each atom row written by exactly one element pass
              g[(size_t)permS[lr] * FPD + col] = c[i];
          }
        }
      }
    }
    __syncthreads();
  }
}

// ---------------------------------------------------------------------------
extern "C" void kernel_launch(void* const* d_in, const int* in_sizes, int n_in,
                              void* d_out, int out_size, void* d_ws, size_t ws_size,
                              hipStream_t stream) {
  const float* FPg  = (const float*)d_in[0];
  const int*   AN   = (const int*)d_in[1];
  const int*   IMG  = (const int*)d_in[2];
  const float* W1   = (const float*)d_in[3];
  const float* B1   = (const float*)d_in[4];
  const float* W2   = (const float*)d_in[5];
  const float* B2   = (const float*)d_in[6];
  const float* W3   = (const float*)d_in[7];
  const float* B3   = (const float*)d_in[8];
  const int*   rows = (const int*)d_in[9];
  const int*   cols = (const int*)d_in[10];
  const float* vals = (const float*)d_in[11];

  float* out = (float*)d_out;            // [600 energy | 180000 forces]
  char*  ws  = (char*)d_ws;
  float* g    = (float*)ws;                                      // 28.8 MB
  int*   perm = (int*)(ws + (size_t)NATOM * FPD * sizeof(float));// 240 KB
  int*   cnt  = perm + NATOM;
  int*   cur  = cnt + 4;

  // zero outputs (energy + forces) and sort counters
  k_zero_f32<<<(NIMG + NFORCEV + 255) / 256, 256, 0, stream>>>(out, NIMG + NFORCEV);
  k_zero_i32<<<1, 32, 0, stream>>>(cnt, 8);
  // counting sort of atoms by element -> contiguous WMMA tiles per element
  k_hist<<<(NATOM + 255) / 256, 256, 0, stream>>>(AN, cnt);
  k_scan<<<1, 1, 0, stream>>>(cnt, cur);
  k_scatter<<<(NATOM + 255) / 256, 256, 0, stream>>>(AN, cur, perm);

  size_t smem = (size_t)(TILE * LDFP + 3 * TILE * LDH + 3 * HID + TILE) * 4 +
                3 * TILE * 4;
  bpnn_mlp<<<dim3(NTILES), 256, smem, stream>>>(FPg, AN, IMG, W1, B1, W2, B2,
                                                W3, B3, perm, g, out);
  k_force<<<(NNZV + 255) / 256, 256, 0, stream>>>(rows, cols, vals, g,
                                                  out + NIMG);
}